// TritonAttention_38826504356236
// MI455X (gfx1250) — compile-verified
//
#include <hip/hip_runtime.h>
#include <hip/hip_bf16.h>

// ---------------------------------------------------------------------------
// Problem constants (match reference)
// ---------------------------------------------------------------------------
#define HQ      32
#define HKV_    8
#define G_      4          // HQ / HKV
#define D_      128
#define HIDDEN_ 4096
#define NQKV    6144       // (HQ + 2*HKV) * D
#define BS_     16         // paged block size
#define MAXB    128        // blocks per sequence
#define BSEQ    64         // batch
#define NBLK    8192       // physical blocks
#define KVROW   (HKV_ * D_)   // 1024 floats between positions in cache

#define LDK     132        // padded LDS row stride for K/V/Q tiles (floats)
#define LDP     18         // padded LDS row stride for P tile (floats)
#define S_      4          // flash-decoding splits per (b, kv)
#define PSTRIDE (G_ * (D_ + 2))   // per-(b,kv,split) partial: m,l,o[128] x 4 rows

typedef __attribute__((ext_vector_type(2))) float v2f;
typedef __attribute__((ext_vector_type(8))) float v8f;

__device__ __forceinline__ v8f wmma_f32(v2f a, v2f b, v8f c) {
    // D = A(16x4) * B(4x16) + C(16x16), fp32, full wave32
    return __builtin_amdgcn_wmma_f32_16x16x4_f32(
        /*neg_a=*/false, a, /*neg_b=*/false, b,
        /*c_mod=*/(short)0, c, /*reuse_a=*/false, /*reuse_b=*/false);
}

__device__ __forceinline__ void wait_ds0() {
    asm volatile("s_wait_dscnt 0" ::: "memory");
}

// Issue one 16x128 fp32 K page + V page into LDS via the async-DMA path.
// 32 x global_load_async_to_lds_b128; per lane: one float4 column per row.
__device__ __forceinline__ void issue_page(const float* __restrict__ kb,
                                           const float* __restrict__ vb,
                                           float* kl, float* vl, int lane) {
    const float* kg = kb + lane * 4;
    const float* vg = vb + lane * 4;
    const unsigned kld = (unsigned)(uintptr_t)(kl + lane * 4);
    const unsigned vld = (unsigned)(uintptr_t)(vl + lane * 4);
    #pragma unroll
    for (int r = 0; r < 16; ++r) {
        asm volatile("global_load_async_to_lds_b128 %0, %1, off"
                     :: "v"(kld + r * (LDK * 4)), "v"(kg + (size_t)r * KVROW)
                     : "memory");
        asm volatile("global_load_async_to_lds_b128 %0, %1, off"
                     :: "v"(vld + r * (LDK * 4)), "v"(vg + (size_t)r * KVROW)
                     : "memory");
    }
}

// ---------------------------------------------------------------------------
// Kernel 1/4: fp32 GEMM C[M,N] = A[M,K] * B[K,N] via V_WMMA_F32_16X16X4_F32.
// ---------------------------------------------------------------------------
__global__ __launch_bounds__(256)
void gemm_f32_wmma(const float* __restrict__ A, const float* __restrict__ B,
                   float* __restrict__ C, int M, int N, int K) {
    const int lane = threadIdx.x & 31;
    const int wave = threadIdx.x >> 5;
    const int ntiles = N >> 4;
    const int mtiles = M >> 4;
    const int tile = blockIdx.x * 8 + wave;
    if (tile >= mtiles * ntiles) return;            // wave-uniform
    const int mt = tile / ntiles, nt = tile % ntiles;
    const int m0 = mt << 4,      n0 = nt << 4;
    const int half = lane >> 4;                     // 0: K=0,1  1: K=2,3
    const int row  = lane & 15;

    const float* ap = A + (size_t)(m0 + row) * K + 2 * half;       // A[m][k]
    const float* bp = B + (size_t)(2 * half) * N + n0 + row;       // B[k][n]
    v8f acc = {};
    #pragma unroll 4
    for (int k0 = 0; k0 < K; k0 += 4) {
        v2f a = *(const v2f*)ap;          // A[m][k+2h], A[m][k+2h+1]
        v2f b = { bp[0], bp[N] };         // B[k+2h][n], B[k+2h+1][n]
        acc = wmma_f32(a, b, acc);
        ap += 4;
        bp += (size_t)4 * N;
    }
    float* cp = C + (size_t)(m0 + 8 * half) * N + n0 + row;
    #pragma unroll
    for (int r = 0; r < 8; ++r) cp[(size_t)r * N] = acc[r];
}

// ---------------------------------------------------------------------------
// Kernel 2/4: paged GQA decode attention, flash-decoding partials.
// grid = (B, HKV, S_), one wave per block, double-buffered async K/V pages.
// ---------------------------------------------------------------------------
__global__ __launch_bounds__(32)
void attn_kernel(const float* __restrict__ qkv,        // [B, NQKV]
                 const float* __restrict__ kcache,     // [NBLK, BS, HKV, D]
                 const float* __restrict__ vcache,     // [NBLK, BS, HKV, D]
                 const int*   __restrict__ seq_lens,   // [B]
                 const int*   __restrict__ btab,       // [B, MAXB]
                 float*       __restrict__ part)       // [B*HKV, S_, PSTRIDE]
{
    const int b    = blockIdx.x;
    const int kv   = blockIdx.y;
    const int sp   = blockIdx.z;
    const int lane = threadIdx.x & 31;
    const int half = lane >> 4;
    const int row  = lane & 15;

    __shared__ float qs[16 * LDK];                 // scaled Q, rows 4..15 zero
    __shared__ float ks[2][16 * LDK];              // double-buffered K page
    __shared__ float vs[2][16 * LDK];              // double-buffered V page
    __shared__ float ps[16 * LDP];                 // P tile, rows 4..15 zero

    const float scale = 0.08838834764831845f;      // D^-0.5

    const int seq       = seq_lens[b];
    const int last      = seq - 1;
    const int ntiles    = (seq + BS_ - 1) / BS_;
    const int last_tile = last / BS_;
    const int last_row  = last % BS_;

    // ---- kick off the first page for this split ---------------------------
    if (sp < ntiles) {
        const int blk = btab[b * MAXB + sp];
        issue_page(kcache + ((size_t)blk * BS_ * HKV_ + kv) * D_,
                   vcache + ((size_t)blk * BS_ * HKV_ + kv) * D_,
                   ks[0], vs[0], lane);
    }

    // ---- stage Q (pre-scaled, zero-padded) + zero P pad (overlaps async) --
    for (int i = lane; i < 16 * D_; i += 32) {
        int r = i >> 7, d = i & (D_ - 1);
        float v = 0.0f;
        if (r < G_) v = qkv[(size_t)b * NQKV + (kv * G_ + r) * D_ + d] * scale;
        qs[r * LDK + d] = v;
    }
    for (int i = lane; i < 16 * LDP; i += 32) ps[i] = 0.0f;

    float m[G_], l[G_];
    v8f   oacc[8];
    #pragma unroll
    for (int g = 0; g < G_; ++g) { m[g] = -__builtin_inff(); l[g] = 0.0f; }
    #pragma unroll
    for (int n = 0; n < 8; ++n) oacc[n] = (v8f){};

    int buf = 0;
    for (int t = sp; t < ntiles; t += S_) {
        // ---- prefetch next page into alternate buffer, wait current -------
        const int tn = t + S_;
        if (tn < ntiles) {
            const int blk2 = btab[b * MAXB + tn];
            issue_page(kcache + ((size_t)blk2 * BS_ * HKV_ + kv) * D_,
                       vcache + ((size_t)blk2 * BS_ * HKV_ + kv) * D_,
                       ks[buf ^ 1], vs[buf ^ 1], lane);
            // async ops retire in order: <=32 left  =>  current page landed
            asm volatile("s_wait_asynccnt 0x20" ::: "memory");
        } else {
            asm volatile("s_wait_asynccnt 0x0" ::: "memory");
        }

        // patch the freshly computed K/V for the new token (cache write)
        if (t == last_tile) {
            const float* knew = qkv + (size_t)b * NQKV + HQ * D_ + kv * D_;
            const float* vnew = qkv + (size_t)b * NQKV + (HQ + HKV_) * D_ + kv * D_;
            for (int d = lane; d < D_; d += 32) {
                ks[buf][last_row * LDK + d] = knew[d];
                vs[buf][last_row * LDK + d] = vnew[d];
            }
        }
        wait_ds0();   // qs/ps staging + patch visible to this wave's ds_loads

        // ---- S = (Q*scale) @ K^T : 32 WMMAs over D ------------------------
        v8f s = {};
        #pragma unroll
        for (int d0 = 0; d0 < D_; d0 += 4) {
            v2f a  = *(const v2f*)&qs[row * LDK + d0 + 2 * half];
            v2f bq = *(const v2f*)&ks[buf][row * LDK + d0 + 2 * half];
            s = wmma_f32(a, bq, s);
        }

        // ---- online softmax per query row ---------------------------------
        const int vcount = min(seq - t * BS_, BS_);
        #pragma unroll
        for (int g = 0; g < G_; ++g) {
            float sv = (lane < 16 && lane < vcount) ? s[g] : -__builtin_inff();
            float tm = sv;
            #pragma unroll
            for (int off = 16; off > 0; off >>= 1)
                tm = fmaxf(tm, __shfl_xor(tm, off, 32));
            const float mnew = fmaxf(m[g], tm);
            const float corr = __expf(m[g] - mnew);
            float p = __expf(sv - mnew);                 // masked lanes -> 0
            float psum = p;
            #pragma unroll
            for (int off = 16; off > 0; off >>= 1)
                psum += __shfl_xor(psum, off, 32);
            l[g] = l[g] * corr + psum;
            m[g] = mnew;
            #pragma unroll
            for (int n = 0; n < 8; ++n) oacc[n][g] *= corr;
            if (lane < 16) ps[g * LDP + lane] = p;
        }
        wait_ds0();   // P tile visible

        // ---- O += P @ V : 8 n-tiles x 4 k-chunks of WMMA ------------------
        #pragma unroll
        for (int n = 0; n < 8; ++n) {
            v8f o = oacc[n];
            #pragma unroll
            for (int p0 = 0; p0 < 16; p0 += 4) {
                v2f a  = *(const v2f*)&ps[row * LDP + p0 + 2 * half];
                v2f bv = { vs[buf][(p0 + 2 * half) * LDK + n * 16 + row],
                           vs[buf][(p0 + 2 * half + 1) * LDK + n * 16 + row] };
                o = wmma_f32(a, bv, o);
            }
            oacc[n] = o;
        }
        wait_ds0();   // this page's ds_loads retire before buffer reuse
        buf ^= 1;
    }

    // ---- write this split's partial (m, l, o) to workspace ----------------
    float* pb = part + ((size_t)(b * HKV_ + kv) * S_ + sp) * PSTRIDE;
    if (lane == 0) {
        #pragma unroll
        for (int g = 0; g < G_; ++g) {
            pb[g * (D_ + 2) + 0] = m[g];
            pb[g * (D_ + 2) + 1] = l[g];
        }
    }
    if (lane < 16) {
        #pragma unroll
        for (int n = 0; n < 8; ++n)
            #pragma unroll
            for (int g = 0; g < G_; ++g)
                pb[g * (D_ + 2) + 2 + n * 16 + lane] = oacc[n][g];
    }
}

// ---------------------------------------------------------------------------
// Kernel 3/4: merge flash-decoding partials -> attention output [B, HQ*D].
// ---------------------------------------------------------------------------
__global__ __launch_bounds__(128)
void attn_merge(const float* __restrict__ part,       // [B*HKV, S_, PSTRIDE]
                float*       __restrict__ attn_out)   // [B, HQ*D]
{
    const int bk = blockIdx.x;                        // b*HKV + kv
    const int b  = bk / HKV_, kv = bk % HKV_;
    const float* pg = part + (size_t)bk * S_ * PSTRIDE;

    for (int e = threadIdx.x; e < G_ * D_; e += 128) {
        const int g = e >> 7, d = e & (D_ - 1);
        const int go = g * (D_ + 2);
        float mg = -__builtin_inff();
        #pragma unroll
        for (int s = 0; s < S_; ++s)
            mg = fmaxf(mg, pg[s * PSTRIDE + go]);
        float Lg = 0.0f, og = 0.0f;
        #pragma unroll
        for (int s = 0; s < S_; ++s) {
            const float c = __expf(pg[s * PSTRIDE + go] - mg);
            Lg += c * pg[s * PSTRIDE + go + 1];
            og += c * pg[s * PSTRIDE + go + 2 + d];
        }
        attn_out[(size_t)b * (HQ * D_) + (kv * G_ + g) * D_ + d] = og / Lg;
    }
}

// ---------------------------------------------------------------------------
// Host-side launch
// ---------------------------------------------------------------------------
extern "C" void kernel_launch(void* const* d_in, const int* in_sizes, int n_in,
                              void* d_out, int out_size, void* d_ws, size_t ws_size,
                              hipStream_t stream) {
    const float* hidden  = (const float*)d_in[0];
    // d_in[1] positions: unused by the reference computation
    const float* kvc     = (const float*)d_in[2];
    // d_in[3] slot_mapping: implied by block_tables + seq_lens (patched in-kernel)
    const int*   seqlens = (const int*)d_in[4];
    const int*   btab    = (const int*)d_in[5];
    const float* wqkv    = (const float*)d_in[6];
    const float* wo      = (const float*)d_in[7];
    float*       out     = (float*)d_out;

    const float* kcache = kvc;
    const float* vcache = kvc + (size_t)NBLK * BS_ * HKV_ * D_;

    float* qkv  = (float*)d_ws;                             // [B, NQKV]
    float* attn = qkv  + (size_t)BSEQ * NQKV;               // [B, HQ*D]
    float* part = attn + (size_t)BSEQ * HQ * D_;            // [B*HKV, S_, PSTRIDE]

    // 1) QKV projection: [64,4096] x [4096,6144]
    {
        const int tiles = (BSEQ / 16) * (NQKV / 16);
        gemm_f32_wmma<<<(tiles + 7) / 8, 256, 0, stream>>>(
            hidden, wqkv, qkv, BSEQ, NQKV, HIDDEN_);
    }
    // 2) paged attention partials (async double-buffered K/V pages)
    {
        dim3 grid(BSEQ, HKV_, S_);
        attn_kernel<<<grid, 32, 0, stream>>>(
            qkv, kcache, vcache, seqlens, btab, part);
    }
    // 3) merge flash-decoding partials
    {
        attn_merge<<<BSEQ * HKV_, 128, 0, stream>>>(part, attn);
    }
    // 4) output projection: [64,4096] x [4096,4096]
    {
        const int tiles = (BSEQ / 16) * (HIDDEN_ / 16);
        gemm_f32_wmma<<<(tiles + 7) / 8, 256, 0, stream>>>(
            attn, wo, out, BSEQ, HIDDEN_, HQ * D_);
    }
}